// CLIP_CPMMC_FSAR_36541581754434
// MI455X (gfx1250) — compile-verified
//
#include <hip/hip_runtime.h>
#include <hip/hip_bf16.h>

// Sizes fixed by the reference setup.
#define Q_  512
#define S_  128
#define T_  16
#define D_  512
#define NROWS_T (Q_ * T_)   // 8192 target frame rows
#define NROWS_S (S_ * T_)   // 2048 support frame rows
#define DISTS_FLOATS ((size_t)Q_ * S_ * T_ * T_)  // 16,777,216 floats = 64 MB

typedef __attribute__((ext_vector_type(2))) float v2f;
typedef __attribute__((ext_vector_type(8))) float v8f;
typedef __attribute__((ext_vector_type(4))) unsigned int v4u;
typedef __attribute__((ext_vector_type(8))) int v8i;
typedef __attribute__((ext_vector_type(4))) int v4i;

__device__ __forceinline__ v2f L2F(const float* p) { return *(const v2f*)p; }

#define WMMA_F32(A, B, C) \
    __builtin_amdgcn_wmma_f32_16x16x4_f32(false, (A), false, (B), (short)0, (C), false, false)

// ---------------------------------------------------------------------------
// Kernel 1: per-frame L2 norms. nrm[0..8191] = target rows, nrm[8192..10239] = support.
// One wave (32 lanes) per 512-long row, butterfly reduce.
// ---------------------------------------------------------------------------
__global__ __launch_bounds__(256) void otam_norms(const float* __restrict__ tgt,
                                                  const float* __restrict__ sup,
                                                  float* __restrict__ nrm) {
    int gw   = (blockIdx.x * blockDim.x + threadIdx.x) >> 5;  // global wave id
    int lane = threadIdx.x & 31;
    if (gw >= NROWS_T + NROWS_S) return;
    const float* src = (gw < NROWS_T) ? (tgt + (size_t)gw * D_)
                                      : (sup + (size_t)(gw - NROWS_T) * D_);
    float s = 0.f;
    for (int k = lane; k < D_; k += 32) {
        float v = src[k];
        s += v * v;
    }
#pragma unroll
    for (int off = 16; off; off >>= 1) s += __shfl_xor(s, off, 32);
    if (lane == 0) nrm[gw] = sqrtf(s);
}

// ---------------------------------------------------------------------------
// Kernel 2: WMMA GEMM. One wave computes FOUR (q,s) 16x16 tiles sharing the
// A operand (target frames of one q).  K = 512 -> 128 WMMA steps per tile.
// Manual depth-2 software pipeline: prefetch operand set for step k+8 while
// issuing the 4 WMMAs of step k, so ~10 loads stay in flight per wave.
// The 4 B tiles are s-consecutive -> one base pointer + immediate offsets
// (j * 8192 floats = 32 KB, within the 24-bit ioffset).
// A 16x4 f32 layout: lane<16 -> (M=lane, K=0..1), lane>=16 -> (M=lane-16, K=2..3);
// B 4x16 symmetric with N = lane%16 -> identical float2 addressing on support.
// Writes dists[q][s][tq][ts] = 1 - D/(||t||*||s|| + 0.01).
// ---------------------------------------------------------------------------
__global__ __launch_bounds__(256) void otam_gemm(const float* __restrict__ tgt,
                                                 const float* __restrict__ sup,
                                                 const float* __restrict__ nrm,
                                                 float* __restrict__ dists) {
    int lane = threadIdx.x & 31;
    int wv   = threadIdx.x >> 5;
    int q    = blockIdx.x;                      // 0..511
    int s0   = (blockIdx.y * 8 + wv) * 4;       // 0,4,...,124
    int row  = lane & 15;                       // M for A, N for B
    int kh   = (lane >> 4) << 1;                // 0 or 2

    const float* At = tgt + (size_t)(q * T_ + row) * D_ + kh;
    const float* Bb = sup + (size_t)(s0 * T_ + row) * D_ + kh;  // +j*8192 for tile j

    v8f acc0 = {}, acc1 = {}, acc2 = {}, acc3 = {};

    // pipeline prologue: steps k=0 (cur) and k=4 (nxt)
    v2f ca  = L2F(At + 0);
    v2f cb0 = L2F(Bb + 0),     cb1 = L2F(Bb + 8192);
    v2f cb2 = L2F(Bb + 16384), cb3 = L2F(Bb + 24576);
    v2f na  = L2F(At + 4);
    v2f nb0 = L2F(Bb + 4),         nb1 = L2F(Bb + 4 + 8192);
    v2f nb2 = L2F(Bb + 4 + 16384), nb3 = L2F(Bb + 4 + 24576);

#pragma unroll 2
    for (int k = 0; k <= 500; k += 4) {   // 126 iterations: compute k, prefetch k+8
        v2f ta  = L2F(At + k + 8);
        v2f tb0 = L2F(Bb + k + 8);
        v2f tb1 = L2F(Bb + k + 8 + 8192);
        v2f tb2 = L2F(Bb + k + 8 + 16384);
        v2f tb3 = L2F(Bb + k + 8 + 24576);
        acc0 = WMMA_F32(ca, cb0, acc0);
        acc1 = WMMA_F32(ca, cb1, acc1);
        acc2 = WMMA_F32(ca, cb2, acc2);
        acc3 = WMMA_F32(ca, cb3, acc3);
        ca = na;  cb0 = nb0; cb1 = nb1; cb2 = nb2; cb3 = nb3;
        na = ta;  nb0 = tb0; nb1 = tb1; nb2 = tb2; nb3 = tb3;
    }
    // epilogue: steps k=504 (cur) and k=508 (nxt)
    acc0 = WMMA_F32(ca, cb0, acc0);
    acc1 = WMMA_F32(ca, cb1, acc1);
    acc2 = WMMA_F32(ca, cb2, acc2);
    acc3 = WMMA_F32(ca, cb3, acc3);
    acc0 = WMMA_F32(na, nb0, acc0);
    acc1 = WMMA_F32(na, nb1, acc1);
    acc2 = WMMA_F32(na, nb2, acc2);
    acc3 = WMMA_F32(na, nb3, acc3);

    float tn[8];
#pragma unroll
    for (int i = 0; i < 8; i++) tn[i] = nrm[q * T_ + i + ((lane >> 4) << 3)];

#pragma unroll
    for (int j = 0; j < 4; j++) {
        int s = s0 + j;
        float sn = nrm[NROWS_T + s * T_ + row];
        size_t base = ((size_t)(q * S_ + s)) * (T_ * T_);
        v8f acc = (j == 0) ? acc0 : (j == 1) ? acc1 : (j == 2) ? acc2 : acc3;
#pragma unroll
        for (int i = 0; i < 8; i++) {
            int tq  = i + ((lane >> 4) << 3);   // C layout: M = i (+8 for hi half)
            float v = 1.f - acc[i] / (tn[i] * sn + 0.01f);
            dists[base + tq * T_ + row] = v;
        }
    }
}

// ---------------------------------------------------------------------------
// Kernel 3: soft-min DTW DP. 128 threads/block, 64 (q,s) tiles staged in LDS.
// Tiles are stride-padded to 260 dwords (bank-conflict-free). Staging uses the
// Tensor Data Mover: one 1-D TENSOR_LOAD_TO_LDS of 16384 dwords with
// pad_interval=7 (every 256 DWORDs) / pad_amount=3 (+4 DWORDs) -> stride 260.
// Lane pair (2p,2p+1) runs the two alignment directions of tile p; results
// combined via shuffle and written once (deterministic across graph replays).
// ---------------------------------------------------------------------------
#define NPAIR 64
#define TSTRIDE 260
#define BIGF 1e30f

__global__ __launch_bounds__(128) void otam_dp(const float* __restrict__ dists,
                                               float* __restrict__ out) {
    __shared__ __align__(16) float lds[NPAIR * TSTRIDE];
    int tid      = threadIdx.x;
    int pairBase = blockIdx.x * NPAIR;

#if __has_builtin(__builtin_amdgcn_tensor_load_to_lds)
    if (tid < 32) {  // wave 0 issues the TDM transfer (EXEC ignored by TDM)
        unsigned long long ga =
            (unsigned long long)(const void*)(dists + (size_t)pairBase * 256);
        unsigned ldsOff = (unsigned)(size_t)(void*)&lds[0];  // wave-relative LDS byte addr

        // D# group 0: count=1 | lds_addr | global_addr[56:0] | type=2
        v4u g0;
        g0.x = 1u;                                   // count=1, user descriptor
        g0.y = ldsOff;                               // lds_addr (bytes)
        g0.z = (unsigned)(ga & 0xFFFFFFFFu);         // global_addr[31:0]
        g0.w = (unsigned)((ga >> 32) & 0x01FFFFFFu) | 0x80000000u;  // addr[56:32] | type=2

        // D# group 1: data_size=4B(2) | pad_enable | pad_interval=7 | pad_amount=3
        //             tensor_dim0=16384, tensor_dim1=1, tile_dim0=16384 (1-D tile)
        //             tensor_dim0_stride=16384
        v8i g1;
        g1[0] = (int)((2u << 16) | (1u << 20) | (7u << 22) | (3u << 25));
        g1[1] = (int)(16384u << 16);   // [15:0] atomic_addr=0, [31:16] tensor_dim0.lo
        g1[2] = (int)(1u << 16);       // [15:0] tensor_dim0.hi=0, [31:16] tensor_dim1.lo=1
        g1[3] = (int)(16384u << 16);   // [15:0] tensor_dim1.hi=0, [31:16] tile_dim0=16384
        g1[4] = 0;                     // tile_dim1=0 (unused), tile_dim2=0
        g1[5] = 16384;                 // tensor_dim0_stride[31:0]
        g1[6] = 0;                     // stride.hi=0, tensor_dim1_stride.lo=0
        g1[7] = 0;
        v4i gz4 = {0, 0, 0, 0};
        v8i gz8 = {0, 0, 0, 0, 0, 0, 0, 0};

        __builtin_amdgcn_tensor_load_to_lds(g0, g1, gz4, gz4, gz8, 0);
        __builtin_amdgcn_s_wait_tensorcnt(0);
    }
#else
    {
        const float4* g = (const float4*)(dists + (size_t)pairBase * 256);
        for (int idx = tid; idx < NPAIR * 64; idx += 128) {
            int p = idx >> 6;
            int e = idx & 63;
            float4 v = g[p * 64 + e];
            *(float4*)&lds[p * TSTRIDE + e * 4] = v;
        }
    }
#endif
    __syncthreads();

    int p   = tid >> 1;
    int dir = tid & 1;
    const float* tile = &lds[p * TSTRIDE];

    // padded DP row: m = 0..17, row l = 0..15. d_pad[l][m] = tile[l][m-1] (m<=16) else 0.
    float prev[18];
    prev[0] = 0.f;
#pragma unroll
    for (int m = 1; m <= 17; m++) {
        float dv = (m <= 16) ? (dir ? tile[(m - 1) * 16 + 0] : tile[0 * 16 + (m - 1)]) : 0.f;
        prev[m] = prev[m - 1] + dv;
    }

    for (int l = 1; l < 16; l++) {
        float left = 0.f;  // cur[m-1]; cur[0] == 0
#pragma unroll
        for (int m = 1; m <= 17; m++) {
            float dv  = (m <= 16) ? (dir ? tile[(m - 1) * 16 + l] : tile[l * 16 + (m - 1)]) : 0.f;
            float pm1 = prev[m - 1];
            float ex  = (m == 1) ? prev[1] : ((m == 17) ? prev[17] : BIGF);
            // softmin(pm1, left, ex) with lambda = 0.5 (stable: shift by min)
            float mn   = fminf(fminf(pm1, left), ex);
            float ssum = __expf((mn - pm1) * 2.f) + __expf((mn - left) * 2.f) +
                         __expf((mn - ex) * 2.f);
            float v = dv + mn - 0.5f * __logf(ssum);
            prev[m - 1] = left;  // finalize new[m-1]; old prev[m-1] is dead now
            left = v;
        }
        prev[17] = left;
    }

    float res = prev[17];
    res += __shfl_xor(res, 1, 32);           // sum the two directions
    if (dir == 0) out[pairBase + p] = res;   // pair index == q*S_ + s
}

// ---------------------------------------------------------------------------
extern "C" void kernel_launch(void* const* d_in, const int* in_sizes, int n_in,
                              void* d_out, int out_size, void* d_ws, size_t ws_size,
                              hipStream_t stream) {
    (void)in_sizes; (void)n_in; (void)out_size; (void)ws_size;
    const float* sup = (const float*)d_in[0];   // support_features [128,16,512]
    const float* tgt = (const float*)d_in[1];   // target_features  [512,16,512]
    float* out   = (float*)d_out;               // [512,128]
    float* dists = (float*)d_ws;                // 64 MB: [Q][S][16][16]
    float* nrm   = dists + DISTS_FLOATS;        // 10240 floats: target then support norms

    // 1. norms: 10240 waves, 8 waves/block
    otam_norms<<<(NROWS_T + NROWS_S + 7) / 8, 256, 0, stream>>>(tgt, sup, nrm);

    // 2. WMMA GEMM: 512 q-blocks x 4 s-groups, 8 waves/block, 4 tiles/wave
    dim3 g2(Q_, S_ / 32);
    otam_gemm<<<g2, 256, 0, stream>>>(tgt, sup, nrm, dists);

    // 3. DP: 65536 pairs / 64 per block
    otam_dp<<<(Q_ * S_) / NPAIR, 128, 0, stream>>>(dists, out);
}